// AdaptiveMessagePassingLayer_5420248727955
// MI455X (gfx1250) — compile-verified
//
#include <hip/hip_runtime.h>

typedef __attribute__((ext_vector_type(2))) float v2f;
typedef __attribute__((ext_vector_type(4))) float v4f;
typedef __attribute__((ext_vector_type(8))) float v8f;

#define D 128          // D_IN == D_OUT == 128
#define CHUNK_ROWS 128 // rows of A staged in LDS per block iteration

// ---------------------------------------------------------------------------
// Kernel 1: W_eff[k][n] = sum_r scales[r] * W[r][k][n]   (128x128 output)
// ---------------------------------------------------------------------------
__global__ void weff_reduce_kernel(const float* __restrict__ W,
                                   const float* __restrict__ s,
                                   float* __restrict__ Weff, int R) {
    int idx = blockIdx.x * blockDim.x + threadIdx.x;
    if (idx < D * D) {
        float acc = 0.0f;
        for (int r = 0; r < R; ++r)
            acc += s[r] * W[r * D * D + idx];
        Weff[idx] = acc;
    }
}

// ---------------------------------------------------------------------------
// Kernel 2: out[N,128] = A[N,128] @ Weff[128,128] via V_WMMA_F32_16X16X4_F32
//
// Block = 256 threads = 8 waves. Wave w owns output columns [w*16, w*16+16).
// B (Weff column slice, 128x16) lives in 32 v2f register fragments per wave.
// A is staged 128 rows at a time into LDS with an XOR swizzle (by row) so the
// per-fragment ds_load_b64 reads are bank-conflict free.
// ---------------------------------------------------------------------------
__global__ __launch_bounds__(256)
void gemm_wmma_f32_kernel(const float* __restrict__ A,
                          const float* __restrict__ Weff,
                          float* __restrict__ out,
                          int Nrows, int nChunks) {
    __shared__ float lds[CHUNK_ROWS * D];   // 64 KB, XOR-swizzled in 4-float blocks

    const int lane  = threadIdx.x & 31;
    const int wave  = threadIdx.x >> 5;     // 0..7 -> output N-tile
    const int lr    = lane & 15;            // 0..15
    const int khalf = (lane >> 4) * 2;      // 0 or 2 (which K pair of the K=4 step)

    // ---- Load B fragments: b[kk] covers K = kk*4 + khalf .. +1, N = wave*16+lr
    // ISA B layout (4x16): lanes 0-15 hold K rows {0,1}, lanes 16-31 hold {2,3}.
    v2f b[32];
    {
        const int ncol = wave * 16 + lr;
        #pragma unroll
        for (int kk = 0; kk < 32; ++kk) {
            b[kk].x = Weff[(kk * 4 + khalf + 0) * D + ncol];
            b[kk].y = Weff[(kk * 4 + khalf + 1) * D + ncol];
        }
    }

    for (int chunk = blockIdx.x; chunk < nChunks; chunk += gridDim.x) {
        const int rowBase = chunk * CHUNK_ROWS;

        __syncthreads();   // previous iteration's LDS reads complete
        // ---- Cooperative staged load: 128 rows x 128 cols, float4, coalesced.
        // LDS layout: row r, 4-float block c -> lds[r*128 + ((c ^ (r & 31)) << 2)]
        #pragma unroll
        for (int i = 0; i < 16; ++i) {
            int slot = i * 256 + threadIdx.x;   // 0..4095
            int r    = slot >> 5;               // 0..127
            int cblk = slot & 31;               // 0..31 (4-float blocks)
            v4f v = {};
            if (rowBase + r < Nrows)
                v = *(const v4f*)(A + (size_t)(rowBase + r) * D + cblk * 4);
            *(v4f*)(&lds[r * D + ((cblk ^ (r & 31)) << 2)]) = v;
        }
        __syncthreads();

        #pragma unroll
        for (int sub = 0; sub < 8; ++sub) {
            const int arow = sub * 16 + lr;        // A row this lane feeds
            const int rsw  = arow & 31;            // swizzle key
            const float* arp = &lds[arow * D];

            v8f c = {};
            #pragma unroll
            for (int kk = 0; kk < 32; ++kk) {
                // A fragment: lane holds (row = lr of tile, K = kk*4 + khalf .. +1)
                v2f a = *(const v2f*)(arp + ((kk ^ rsw) << 2) + khalf);
                c = __builtin_amdgcn_wmma_f32_16x16x4_f32(
                        /*neg_a=*/false, a, /*neg_b=*/false, b[kk],
                        /*c_mod=*/(short)0, c, /*reuse_a=*/false, /*reuse_b=*/false);
            }

            // ---- Store 16x16 C tile: VGPR r8 holds rows r8 (lanes 0-15) / r8+8
            const int col = wave * 16 + lr;
            #pragma unroll
            for (int r8 = 0; r8 < 8; ++r8) {
                int row = rowBase + sub * 16 + r8 + (lane >> 4) * 8;
                if (row < Nrows)
                    out[(size_t)row * D + col] = c[r8];
            }
        }
    }
}

// ---------------------------------------------------------------------------
extern "C" void kernel_launch(void* const* d_in, const int* in_sizes, int n_in,
                              void* d_out, int out_size, void* d_ws, size_t ws_size,
                              hipStream_t stream) {
    const float* inputs = (const float*)d_in[0];   // [N, 128]
    const float* relW   = (const float*)d_in[1];   // [R, 128, 128]
    const float* relS   = (const float*)d_in[2];   // [R]
    float* out  = (float*)d_out;
    float* Weff = (float*)d_ws;                    // 128*128 floats = 64 KB scratch

    const int Nrows = in_sizes[0] / D;
    const int R     = in_sizes[2];

    // Fold relation matrices into a single effective weight matrix.
    weff_reduce_kernel<<<(D * D + 255) / 256, 256, 0, stream>>>(relW, relS, Weff, R);

    // Main GEMM.
    const int nChunks = (Nrows + CHUNK_ROWS - 1) / CHUNK_ROWS;
    int grid = nChunks < 2048 ? nChunks : 2048;
    gemm_wmma_f32_kernel<<<grid, 256, 0, stream>>>(inputs, Weff, out, Nrows, nChunks);
}